// ContextPromptGenerator_70514773066341
// MI455X (gfx1250) — compile-verified
//
#include <hip/hip_runtime.h>
#include <math.h>

// Problem sizes (fixed by the reference)
#define B_ 16
#define S_ 2048
#define C_ 512
#define H_ 4096
#define D_ 1024
#define P_ 32

typedef __attribute__((ext_vector_type(2))) float v2f;
typedef __attribute__((ext_vector_type(8))) float v8f;

// ---------------------------------------------------------------------------
// Kernel 1: ragged mean of gathered embedding rows -> ctx_mean [B,H]
// grid (B, H/1024), block 256; each thread owns one float4 column chunk.
// ---------------------------------------------------------------------------
__global__ void ctx_mean_kernel(const int* __restrict__ ids,
                                const int* __restrict__ clen,
                                const float* __restrict__ embed,
                                float* __restrict__ ctx_mean) {
    int b  = blockIdx.x;
    int h4 = blockIdx.y * blockDim.x + threadIdx.x;   // float4 index into H/4
    int n  = clen[b];
    if (n < 1) n = 1;
    if (n > C_) n = C_;
    const int* idrow = ids + b * C_;
    float ax = 0.f, ay = 0.f, az = 0.f, aw = 0.f;
    for (int c = 0; c < n; ++c) {
        int id = idrow[c];
        const float4* row = (const float4*)(embed + (size_t)id * H_);
        float4 v = row[h4];
        ax += v.x; ay += v.y; az += v.z; aw += v.w;
    }
    float inv = 1.0f / (float)n;
    float4 r; r.x = ax * inv; r.y = ay * inv; r.z = az * inv; r.w = aw * inv;
    ((float4*)(ctx_mean + (size_t)b * H_))[h4] = r;
}

// ---------------------------------------------------------------------------
// Kernel 2: per-(batch,bin) sum of hidden rows -> binsum [B*P, H]
// Bin t-range in "seq" space is [start,end); hidden rows are t in
// [max(start,1),end), hidden index t-1. grid (B*P, H/1024), block 256.
// This is the dominant 536 MB HBM stream.
// ---------------------------------------------------------------------------
__global__ void binsum_kernel(const float* __restrict__ hs,
                              const int* __restrict__ slen,
                              float* __restrict__ binsum) {
    int bj = blockIdx.x;              // 0..B*P-1
    int b  = bj / P_;
    int j  = bj % P_;
    int h4 = blockIdx.y * blockDim.x + threadIdx.x;
    int L     = slen[b] + 1;                       // valid rows incl. ctx token
    int start = (j * L) / P_;
    int end   = ((j + 1) * L + P_ - 1) / P_;
    int sh    = start < 1 ? 1 : start;             // first hidden seq index
    float ax = 0.f, ay = 0.f, az = 0.f, aw = 0.f;
    const float4* base = (const float4*)(hs + (size_t)b * S_ * H_);
    #pragma unroll 4
    for (int t = sh; t < end; ++t) {
        float4 v = base[(size_t)(t - 1) * (H_ / 4) + h4];
        ax += v.x; ay += v.y; az += v.z; aw += v.w;
    }
    float4 r; r.x = ax; r.y = ay; r.z = az; r.w = aw;
    ((float4*)binsum)[(size_t)bj * (H_ / 4) + h4] = r;
}

// ---------------------------------------------------------------------------
// LDS-staged double-buffered fp32 GEMM via V_WMMA_F32_16X16X4_F32.
//
// Block = 256 threads = 8 waves. Macro tile 64(M) x 128(N); wave grid 4x2,
// each wave computes a 16x64 strip = NT=4 WMMA tiles (one A frag reused).
// K is chunked by KB=16; chunk c+1 is loaded coalesced (global b128) into
// registers while chunk c is computed out of LDS (ds_load_b64/b32); one
// s_barrier per chunk. Split-K: each block covers K/sk_count, partials go
// to Part[sk][M][N]; sk_count==1 writes Dst (+bias) directly.
//
// A-frag layout (ISA 7.12.2): lanes 0-15 hold K=k+0,k+1; lanes 16-31 K=k+2,k+3.
// D layout: VGPR v -> row m0 + v + 8*(lane>=16), col n0 + (lane&15).
// Requires N%128==0, (K/sk_count)%16==0. M may be < 64 (guarded).
// ---------------------------------------------------------------------------
#define MB 64
#define NB 128
#define KB 16
#define KBP 20   // padded A row stride (floats): 80B keeps float4 align, kills bank conflicts
#define NT 4

__global__ void gemm_wmma_lds(const float* __restrict__ A,
                              const float* __restrict__ Bm,
                              const float* __restrict__ bias,
                              float* __restrict__ Dst,
                              float* __restrict__ Part,
                              int M, int N, int K, int sk_count) {
    __shared__ float As[2][MB * KBP];   // 2 * 64*20*4 = 10.0 KB
    __shared__ float Bs[2][KB * NB];    // 2 * 16*128*4 = 16.0 KB

    int tilesN  = N / NB;
    int numMblk = (M + MB - 1) / MB;
    int bid  = blockIdx.x;
    int nblk = bid % tilesN;
    int tmp  = bid / tilesN;
    int mblk = tmp % numMblk;
    int sk   = tmp / numMblk;
    int Kper = K / sk_count;
    int kbeg = sk * Kper;

    int tid  = threadIdx.x;
    int wave = tid >> 5;
    int lane = tid & 31;
    int l15  = lane & 15;
    int khalf = (lane >> 4) << 1;      // 0 for lanes 0-15, 2 for lanes 16-31
    int mloc = (wave >> 1) * 16;       // wave row within macro tile
    int nloc = (wave & 1) * 64;        // wave col within macro tile

    // cooperative A load: thread -> one float4. row a_r=tid/4, float col a_c=(tid%4)*4
    int a_r = tid >> 2;
    int a_c = (tid & 3) * 4;
    int a_grow = mblk * MB + a_r;
    bool a_ok  = a_grow < M;
    const float* a_gptr = A + (size_t)(a_ok ? a_grow : 0) * K + a_c;

    // cooperative B load: thread -> two float4s (512 float4 total)
    int b_r0 = tid >> 5;               // rows 0..7
    int b_c0 = (tid & 31) * 4;         // 128 floats per row
    const float* b_gptr = Bm + (size_t)nblk * NB;

    v8f acc[NT] = {};
    const float4 f4z = make_float4(0.f, 0.f, 0.f, 0.f);

    // ---- prologue: chunk 0 -> LDS buffer 0
    {
        float4 av = a_ok ? *(const float4*)(a_gptr + kbeg) : f4z;
        *(float4*)&As[0][a_r * KBP + a_c] = av;
        float4 bv0 = *(const float4*)(b_gptr + (size_t)(kbeg + b_r0) * N + b_c0);
        float4 bv1 = *(const float4*)(b_gptr + (size_t)(kbeg + b_r0 + 8) * N + b_c0);
        *(float4*)&Bs[0][b_r0 * NB + b_c0] = bv0;
        *(float4*)&Bs[0][(b_r0 + 8) * NB + b_c0] = bv1;
    }
    __syncthreads();

    int nchunks = Kper / KB;
    int buf = 0;
    for (int c = 0; c < nchunks; ++c) {
        // prefetch next chunk into registers (overlaps with compute below)
        float4 av = f4z, bv0 = f4z, bv1 = f4z;
        bool have_next = (c + 1 < nchunks);
        if (have_next) {
            int kc = kbeg + (c + 1) * KB;
            if (a_ok) av = *(const float4*)(a_gptr + kc);
            bv0 = *(const float4*)(b_gptr + (size_t)(kc + b_r0) * N + b_c0);
            bv1 = *(const float4*)(b_gptr + (size_t)(kc + b_r0 + 8) * N + b_c0);
        }

        // compute chunk c out of LDS buffer buf
        #pragma unroll
        for (int kk = 0; kk < KB; kk += 4) {
            v2f a = *(const v2f*)&As[buf][(mloc + l15) * KBP + kk + khalf];
            #pragma unroll
            for (int i = 0; i < NT; ++i) {
                v2f bf;
                bf.x = Bs[buf][(kk + khalf) * NB + nloc + i * 16 + l15];
                bf.y = Bs[buf][(kk + khalf + 1) * NB + nloc + i * 16 + l15];
                acc[i] = __builtin_amdgcn_wmma_f32_16x16x4_f32(
                    /*neg_a=*/false, a, /*neg_b=*/false, bf,
                    /*c_mod=*/(short)0, acc[i],
                    /*reuse_a=*/false, /*reuse_b=*/false);
            }
        }

        if (have_next) {
            // buf^1 was last read in iteration c-1 (barrier already passed)
            *(float4*)&As[buf ^ 1][a_r * KBP + a_c] = av;
            *(float4*)&Bs[buf ^ 1][b_r0 * NB + b_c0] = bv0;
            *(float4*)&Bs[buf ^ 1][(b_r0 + 8) * NB + b_c0] = bv1;
            __syncthreads();
            buf ^= 1;
        }
    }

    // ---- epilogue
    int m0g = mblk * MB + mloc + ((lane >> 4) * 8);
    int n0g = nblk * NB + nloc + l15;
    if (sk_count == 1) {
        #pragma unroll
        for (int i = 0; i < NT; ++i) {
            int col = n0g + i * 16;
            float bv = bias ? bias[col] : 0.0f;
            #pragma unroll
            for (int v = 0; v < 8; ++v) {
                int row = m0g + v;
                if (row < M) Dst[(size_t)row * N + col] = acc[i][v] + bv;
            }
        }
    } else {
        float* pp = Part + (size_t)sk * M * N;
        #pragma unroll
        for (int i = 0; i < NT; ++i) {
            int col = n0g + i * 16;
            #pragma unroll
            for (int v = 0; v < 8; ++v) {
                int row = m0g + v;
                if (row < M) pp[(size_t)row * N + col] = acc[i][v];
            }
        }
    }
}

// ---------------------------------------------------------------------------
// Split-K reduction: Dst[idx] = sum_sk Part[sk][idx] (+ bias[idx % N]).
// Fixed summation order -> bitwise deterministic.
// ---------------------------------------------------------------------------
__global__ void ksplit_reduce(const float* __restrict__ Part,
                              const float* __restrict__ bias,
                              float* __restrict__ Dst,
                              int MN, int N, int sk_count) {
    int idx = blockIdx.x * blockDim.x + threadIdx.x;
    if (idx >= MN) return;
    float s = 0.f;
    for (int k = 0; k < sk_count; ++k) s += Part[(size_t)k * MN + idx];
    if (bias) s += bias[idx % N];
    Dst[idx] = s;
}

// ---------------------------------------------------------------------------
// Kernel 4: rebuild pooled value from bin sums + biases + ctx term, apply silu.
// pooled = (Yd + n_h*bd + [start==0]*(Yc+bc)) / n_tot ; Z = pooled*sigmoid(pooled)
// ---------------------------------------------------------------------------
__global__ void pooled_silu_kernel(const float* __restrict__ Yd,
                                   const float* __restrict__ Yc,
                                   const float* __restrict__ bc,
                                   const float* __restrict__ bd,
                                   const int* __restrict__ slen,
                                   float* __restrict__ Z) {
    int idx = blockIdx.x * blockDim.x + threadIdx.x;   // over B*P*D
    int d   = idx % D_;
    int bj  = idx / D_;
    int b   = bj / P_;
    int j   = bj % P_;
    int L     = slen[b] + 1;
    int start = (j * L) / P_;
    int end   = ((j + 1) * L + P_ - 1) / P_;
    int ntot  = end - start;
    int sh    = start < 1 ? 1 : start;
    int nh    = end - sh;
    float val = Yd[idx] + (float)nh * bd[d];
    if (start == 0) val += Yc[b * D_ + d] + bc[d];
    float pooled = val / (float)ntot;
    float sig = 1.0f / (1.0f + expf(-pooled));
    Z[idx] = pooled * sig;
}

// ---------------------------------------------------------------------------
// Launch
// ---------------------------------------------------------------------------
extern "C" void kernel_launch(void* const* d_in, const int* in_sizes, int n_in,
                              void* d_out, int out_size, void* d_ws, size_t ws_size,
                              hipStream_t stream) {
    (void)in_sizes; (void)n_in; (void)out_size; (void)ws_size;

    const int*   context_ids = (const int*)  d_in[0];   // [B,C]
    const float* hidden      = (const float*)d_in[1];   // [B,S,H]
    const int*   seq_len     = (const int*)  d_in[2];   // [B]
    const int*   ctx_len     = (const int*)  d_in[3];   // [B]
    const float* embed       = (const float*)d_in[4];   // [V,H]
    const float* Wc          = (const float*)d_in[5];   // [H,D]
    const float* bc          = (const float*)d_in[6];   // [D]
    const float* Wd          = (const float*)d_in[7];   // [H,D]
    const float* bd          = (const float*)d_in[8];   // [D]
    const float* Wu          = (const float*)d_in[9];   // [D,H]
    const float* bu          = (const float*)d_in[10];  // [H]
    float*       out         = (float*)d_out;           // [B,P,H]

    float* ws       = (float*)d_ws;
    float* ctx_mean = ws;                         // B*H      =   65536
    float* binsum   = ctx_mean + B_ * H_;         // B*P*H    = 2097152
    float* Yc       = binsum + B_ * P_ * H_;      // B*D      =   16384
    float* Yd       = Yc + B_ * D_;               // B*P*D    =  524288
    float* Z        = Yd + B_ * P_ * D_;          // B*P*D    =  524288
    float* part     = Z + B_ * P_ * D_;           // max 4*512*1024 = 2097152
    // total ~21.3 MB of workspace

    // 1) ragged mean of context embeddings
    ctx_mean_kernel<<<dim3(B_, H_ / 1024), 256, 0, stream>>>(
        context_ids, ctx_len, embed, ctx_mean);

    // 2) per-bin hidden row sums (dominant 536 MB stream)
    binsum_kernel<<<dim3(B_ * P_, H_ / 1024), 256, 0, stream>>>(
        hidden, seq_len, binsum);

    // 3a) Yd = binsum @ Wd : M=512,N=1024,K=4096, SK=4
    //     blocks = SK * (M/64) * (N/128) = 4*8*8 = 256
    gemm_wmma_lds<<<256, 256, 0, stream>>>(binsum, Wd, nullptr, Yd, part,
                                           B_ * P_, D_, H_, 4);
    ksplit_reduce<<<(B_ * P_ * D_) / 256, 256, 0, stream>>>(
        part, nullptr, Yd, B_ * P_ * D_, D_, 4);

    // 3b) Yc = ctx_mean @ Wc : M=16,N=1024,K=4096, SK=8
    //     blocks = 8 * 1 * 8 = 64 (M guarded inside kernel)
    gemm_wmma_lds<<<64, 256, 0, stream>>>(ctx_mean, Wc, nullptr, Yc, part,
                                          B_, D_, H_, 8);
    ksplit_reduce<<<(B_ * D_) / 256, 256, 0, stream>>>(
        part, nullptr, Yc, B_ * D_, D_, 8);

    // 4) pooled + silu
    pooled_silu_kernel<<<(B_ * P_ * D_) / 256, 256, 0, stream>>>(
        Yd, Yc, bc, bd, seq_len, Z);

    // 5) out = Z @ Wu + bu : M=512,N=4096,K=1024, SK=1 (fused bias)
    //     blocks = 1 * 8 * 32 = 256
    gemm_wmma_lds<<<256, 256, 0, stream>>>(Z, Wu, bu, out, part,
                                           B_ * P_, H_, D_, 1);
}